// CloudSSM_25572235280789
// MI455X (gfx1250) — compile-verified
//
#include <hip/hip_runtime.h>
#include <math.h>

// ---------------- problem constants (match reference) ----------------
#define BSZ   32
#define DCH   96
#define HH    64
#define WW    64
#define LL    (HH * WW)          // 4096
#define NST   4
#define RNK   6
#define KPROJ 14                 // R + 2N
#define LOG_EPS_F (-20.7232658f) // log(1e-9)

typedef __attribute__((ext_vector_type(2))) float v2f;
typedef __attribute__((ext_vector_type(8))) float v8f;

// =====================================================================
// Kernel 1: depthwise 5x5 conv + bias + SiLU  ->  xs (B, D, L) in ws
// one block = (b, d, 4-row strip); LDS tile 8 x 68 with halo
// =====================================================================
__global__ void conv_silu_kernel(const float* __restrict__ x,
                                 const float* __restrict__ conv_w,
                                 const float* __restrict__ conv_b,
                                 float* __restrict__ xs) {
    const int bid  = blockIdx.x;
    const int tile = bid & 15;            // H/4 = 16 strips
    const int d    = (bid >> 4) % DCH;
    const int b    = bid / (16 * DCH);
    const int h0   = tile * 4;

    __shared__ float t[8][68];

    const int tid = threadIdx.x;          // 256 threads
    const float* xin = x + ((size_t)b * DCH + d) * LL;

    // cooperative halo load (8 rows x 68 cols = 544 elems)
    for (int i = tid; i < 8 * 68; i += 256) {
        const int r  = i / 68;
        const int c  = i % 68;
        const int hh = h0 - 2 + r;
        const int wc = c - 2;
        float v = 0.0f;
        if (hh >= 0 && hh < HH && wc >= 0 && wc < WW)
            v = xin[hh * WW + wc];
        t[r][c] = v;
    }
    __syncthreads();

    // weights for this channel (block-uniform -> scalar loads)
    float wv[25];
#pragma unroll
    for (int j = 0; j < 25; ++j) wv[j] = conv_w[d * 25 + j];
    const float bv = conv_b[d];

    const int ty = tid >> 6;              // 0..3
    const int tx = tid & 63;              // 0..63

    float acc = bv;
#pragma unroll
    for (int ky = 0; ky < 5; ++ky)
#pragma unroll
        for (int kx = 0; kx < 5; ++kx)
            acc = fmaf(t[ty + ky][tx + kx], wv[ky * 5 + kx], acc);

    // SiLU
    const float s = acc / (1.0f + expf(-acc));
    xs[((size_t)b * DCH + d) * LL + (size_t)(h0 + ty) * WW + tx] = s;
}

// =====================================================================
// Kernel 2: x_dbl = x_proj_w (14x96, zero-padded to 16x96) @ xs (96 x L)
// per-batch skinny GEMM via V_WMMA_F32_16X16X4_F32 (wave32).
// Pipeline: (1) stage padded A in LDS (6 KB, no predication on hot path),
//           (2) issue ALL 48 B-matrix global loads into registers,
//           (3) run the 24-WMMA accumulation chain fed by ds_load_b64.
// Layouts (f32 16x16x4):
//   A: lane m=lane&15 -> M=m;  VGPR0 = K 2*half, VGPR1 = K 2*half+1
//   B: lane n=lane&15 -> N=n;  VGPR0 = K 2*half, VGPR1 = K 2*half+1
//   C/D: lane n -> col n; VGPR i -> row i + 8*half
// =====================================================================
__global__ void proj_wmma_kernel(const float* __restrict__ xs,
                                 const float* __restrict__ x_proj_w,
                                 float* __restrict__ xdbl) {
    __shared__ float wA[16][DCH];         // zero-padded A, 6 KB

    const int tid = threadIdx.x;          // 256 threads = 8 waves
    // cooperative staging of padded A (rows 14,15 = 0)
    for (int i = tid; i < 16 * DCH; i += 256) {
        const int row = i / DCH;
        const int k   = i % DCH;
        wA[row][k] = (row < KPROJ) ? x_proj_w[row * DCH + k] : 0.0f;
    }
    __syncthreads();

    const int wid  = (blockIdx.x * blockDim.x + tid) >> 5;
    const int lane = tid & 31;
    const int tcol = wid & 255;           // L/16 = 256 tiles
    const int b    = wid >> 8;
    const int l0   = tcol * 16;

    const int half = lane >> 4;
    const int m    = lane & 15;

    const float* xb = xs + (size_t)b * DCH * LL + (l0 + m);

    // ---- phase 1: all B loads in flight (clauseable, hides latency) ----
    v2f breg[DCH / 4];
#pragma unroll
    for (int kk = 0; kk < DCH / 4; ++kk) {
        const int k0 = kk * 4 + 2 * half;
        breg[kk].x = xb[(size_t)k0 * LL];
        breg[kk].y = xb[(size_t)(k0 + 1) * LL];
    }

    // ---- phase 2: 24-deep WMMA accumulation chain, A from LDS ----
    v8f acc = {0.f, 0.f, 0.f, 0.f, 0.f, 0.f, 0.f, 0.f};
#pragma unroll
    for (int kk = 0; kk < DCH / 4; ++kk) {
        const int k0 = kk * 4 + 2 * half;
        const v2f a = *(const v2f*)&wA[m][k0];   // 8B-aligned ds_load_b64
        acc = __builtin_amdgcn_wmma_f32_16x16x4_f32(
                  false, a, false, breg[kk], (short)0, acc, false, false);
    }

    // scatter D: row = i + 8*half, col = l0 + m; keep rows 0..13
#pragma unroll
    for (int i = 0; i < 8; ++i) {
        const int row = i + 8 * half;
        if (row < KPROJ)
            xdbl[((size_t)b * KPROJ + row) * LL + l0 + m] = acc[i];
    }
}

// =====================================================================
// Kernel 3: selective scan. One wave32 per (b,d) channel; 128 chunks
// of 32 along L; Hillis-Steele inclusive prefix sums with carries for
// both coupled cumsums (log-decay S1, compensated numerator S2).
// delta is recomputed inline from dts rows (6 FMAs/element).
// =====================================================================
__device__ __forceinline__ float wave_incl_sum(float v, int lane) {
#pragma unroll
    for (int off = 1; off < 32; off <<= 1) {
        const float t = __shfl_up(v, off, 32);
        if (lane >= off) v += t;
    }
    return v;
}

__global__ void scan_kernel(const float* __restrict__ xs,
                            const float* __restrict__ xdbl,
                            const float* __restrict__ dt_w,
                            const float* __restrict__ dt_b,
                            const float* __restrict__ A_logs,
                            const float* __restrict__ Ds,
                            float* __restrict__ out) {
    const int wid  = (blockIdx.x * blockDim.x + threadIdx.x) >> 5;
    const int lane = threadIdx.x & 31;
    const int d    = wid % DCH;
    const int b    = wid / DCH;

    float wr[RNK];
#pragma unroll
    for (int r = 0; r < RNK; ++r) wr[r] = dt_w[d * RNK + r];
    const float dtb = dt_b[d];
    const float Dd  = Ds[d];
    float An[NST];
#pragma unroll
    for (int n = 0; n < NST; ++n) An[n] = -expf(A_logs[d * NST + n]);

    const float* xch = xs   + ((size_t)b * DCH + d) * LL;
    const float* xdb = xdbl + (size_t)b * KPROJ * LL;
    float*       och = out  + ((size_t)b * DCH + d) * LL;

    float c1[NST] = {0.f, 0.f, 0.f, 0.f};   // running cumsum(log_dA)
    float c2[NST] = {0.f, 0.f, 0.f, 0.f};   // running cumsum(dB_u / P)

    for (int c = 0; c < LL / 32; ++c) {
        const int l = c * 32 + lane;
        const float xv = xch[l];

        // delta = softplus( dt_w[d,:] . dts[b,:,l] + dt_b[d] )
        float ad = dtb;
#pragma unroll
        for (int r = 0; r < RNK; ++r)
            ad = fmaf(wr[r], xdb[(size_t)r * LL + l], ad);
        const float delta = (ad > 20.0f) ? ad : log1pf(expf(ad));

        float y = xv * Dd;

#pragma unroll
        for (int n = 0; n < NST; ++n) {
            const float Bn = xdb[(size_t)(RNK + n) * LL + l];
            const float Cn = xdb[(size_t)(RNK + NST + n) * LL + l];

            const float lda = fmaxf(delta * An[n], LOG_EPS_F);
            float s1 = wave_incl_sum(lda, lane) + c1[n];
            const float P = expf(s1);

            const float q  = (delta * Bn * xv) / fmaxf(P, 1e-9f);
            float s2 = wave_incl_sum(q, lane) + c2[n];

            y = fmaf(P * s2, Cn, y);

            c1[n] = __shfl(s1, 31, 32);
            c2[n] = __shfl(s2, 31, 32);
        }
        och[l] = y;
    }
}

// =====================================================================
// launch
// inputs: 0:x 1:conv_w 2:conv_b 3:x_proj_w 4:dt_w 5:dt_b 6:A_logs 7:Ds
// ws layout: xs (B*D*L f32, 50.3MB) | xdbl (B*14*L f32, 7.3MB)  ~55MB
// =====================================================================
extern "C" void kernel_launch(void* const* d_in, const int* in_sizes, int n_in,
                              void* d_out, int out_size, void* d_ws, size_t ws_size,
                              hipStream_t stream) {
    (void)in_sizes; (void)n_in; (void)out_size; (void)ws_size;

    const float* x        = (const float*)d_in[0];
    const float* conv_w   = (const float*)d_in[1];
    const float* conv_b   = (const float*)d_in[2];
    const float* x_proj_w = (const float*)d_in[3];
    const float* dt_w     = (const float*)d_in[4];
    const float* dt_b     = (const float*)d_in[5];
    const float* A_logs   = (const float*)d_in[6];
    const float* Ds       = (const float*)d_in[7];
    float* out = (float*)d_out;

    float* xs   = (float*)d_ws;
    float* xdbl = xs + (size_t)BSZ * DCH * LL;

    // 1) depthwise conv + SiLU
    {
        dim3 grid(BSZ * DCH * (HH / 4));
        conv_silu_kernel<<<grid, 256, 0, stream>>>(x, conv_w, conv_b, xs);
    }
    // 2) projection GEMM via WMMA: B*(L/16) waves, 8 waves/block
    {
        const int nwaves = BSZ * (LL / 16);           // 8192
        dim3 grid(nwaves / 8);
        proj_wmma_kernel<<<grid, 256, 0, stream>>>(xs, x_proj_w, xdbl);
    }
    // 3) selective scan: one wave per (b,d) channel, 8 waves/block
    {
        const int nwaves = BSZ * DCH;                 // 3072
        dim3 grid(nwaves / 8);
        scan_kernel<<<grid, 256, 0, stream>>>(xs, xdbl, dt_w, dt_b,
                                              A_logs, Ds, out);
    }
}